// DynamicModel_28484223107437
// MI455X (gfx1250) — compile-verified
//
#include <hip/hip_runtime.h>
#include <cstdint>

#define TPB  256
#define SPT  2                        // samples per thread per tile
#define TILE (TPB * SPT)              // 512 samples / tile
#define TILE_BYTES (TILE * 32)        // 16 KB / tile
#define CHUNKS_PER_THREAD (TILE_BYTES / 16 / TPB)   // 4 x b128 per thread

#if defined(__AMDGCN__) && __has_builtin(__builtin_amdgcn_global_load_async_to_lds_b128)
#define USE_ASYNC 1
#else
#define USE_ASYNC 0
#endif

#if USE_ASYNC
typedef int v4i __attribute__((vector_size(4 * sizeof(int))));
typedef __attribute__((address_space(1))) v4i* gptr_v4i;
typedef __attribute__((address_space(3))) v4i* lptr_v4i;

__device__ __forceinline__ void wait_async0() {
#if __has_builtin(__builtin_amdgcn_s_wait_asynccnt)
    __builtin_amdgcn_s_wait_asynccnt(0);
#else
    asm volatile("s_wait_asynccnt 0x0" ::: "memory");
#endif
}
#endif

__device__ __forceinline__ float rcp_fast(float v) {
#if defined(__AMDGCN__)
    return __builtin_amdgcn_rcpf(v);   // v_rcp_f32
#else
    return 1.0f / v;
#endif
}

// atan on [0,1] via odd minimax polynomial (Hastings), ~1e-5 max err
__device__ __forceinline__ float atan_poly(float t) {
    float t2 = t * t;
    float p = fmaf(t2, 0.0208351f, -0.0851330f);
    p = fmaf(t2, p, 0.1801410f);
    p = fmaf(t2, p, -0.3302995f);
    p = fmaf(t2, p, 0.9998660f);
    return t * p;
}

__device__ __forceinline__ float fast_atan2f(float y, float x) {
    float ax = __builtin_fabsf(x), ay = __builtin_fabsf(y);
    float mx = fmaxf(ax, ay), mn = fminf(ax, ay);
    float t = mn * rcp_fast(mx);
    float r = atan_poly(t);
    r = (ay > ax) ? (1.57079632679f - r) : r;
    r = (x < 0.0f) ? (3.14159265359f - r) : r;
    return __builtin_copysignf(r, y);
}

struct F3 { float a, b, c; };

__global__ __launch_bounds__(TPB)
void dyn_kernel(const float* __restrict__ x,
                const float* __restrict__ fy_w1, const float* __restrict__ fy_b1,
                const float* __restrict__ fy_w2, const float* __restrict__ fy_b2,
                const float* __restrict__ ry_w1, const float* __restrict__ ry_b1,
                const float* __restrict__ ry_w2, const float* __restrict__ ry_b2,
                const float* __restrict__ rx_wp, const float* __restrict__ rx_bp,
                float* __restrict__ out, int B, int tiles_per_block)
{
    __shared__ float lds[2][TILE * 8];
    const int tid = threadIdx.x;

    // Weights: uniform addresses -> scalar loads, values live in SGPRs
    float w1f[12], b1f[12], w2f[12], w1r[12], b1r[12], w2r[12];
#pragma unroll
    for (int i = 0; i < 12; ++i) {
        w1f[i] = fy_w1[i]; b1f[i] = fy_b1[i]; w2f[i] = fy_w2[i];
        w1r[i] = ry_w1[i]; b1r[i] = ry_b1[i]; w2r[i] = ry_w2[i];
    }
    const float fb2 = fy_b2[0], rb2 = ry_b2[0], rxw = rx_wp[0], rxb = rx_bp[0];

    const long firstTile = (long)blockIdx.x * tiles_per_block;
    const long maxByte   = (long)B * 32 - 16;

    auto issue = [&](long t, int buf) {
        long gbase = (firstTile + t) * (long)TILE_BYTES;
#if USE_ASYNC
#pragma unroll
        for (int k = 0; k < CHUNKS_PER_THREAD; ++k) {
            long c  = tid + k * TPB;
            long gb = gbase + c * 16; if (gb > maxByte) gb = maxByte;
            const char* gp = (const char*)x + gb;
            char*       lp = (char*)&lds[buf][0] + c * 16;
            __builtin_amdgcn_global_load_async_to_lds_b128(
                (gptr_v4i)(uintptr_t)gp,
                (lptr_v4i)lp,
                0, 0);
        }
#else
#pragma unroll
        for (int k = 0; k < CHUNKS_PER_THREAD; ++k) {
            long c  = tid + k * TPB;
            long gb = gbase + c * 16; if (gb > maxByte) gb = maxByte;
            float4 v = *(const float4*)((const char*)x + gb);
            *(float4*)((char*)&lds[buf][0] + c * 16) = v;
        }
#endif
    };

    issue(0, 0);

    for (int t = 0; t < tiles_per_block; ++t) {
        const int buf = t & 1;
#if USE_ASYNC
        wait_async0();
#endif
        __syncthreads();                         // tile t fully in LDS (all waves)
        if (t + 1 < tiles_per_block) issue(t + 1, buf ^ 1);   // overlap next load

        const long sbase = (firstTile + t) * (long)TILE;
#pragma unroll
        for (int k = 0; k < SPT; ++k) {
            const int  ls = tid + k * TPB;
            const long s  = sbase + ls;
            const float4 lo = *(const float4*)&lds[buf][ls * 8];
            const float4 hi = *(const float4*)&lds[buf][ls * 8 + 4];
            const float pwm = lo.x, theta = lo.z, bvx = lo.w;
            const float bvy = hi.x, bw = hi.y, pitch = hi.w;

            const float st = __sinf(theta), ct = __cosf(theta), sp = __sinf(pitch);
            const float C_DRIVE = 3.45f * 0.919f / (0.34f * 1265.0f);
            const float a_pred  = (pwm > 0.0f ? 550.0f : 650.0f) * pwm * C_DRIVE;

            float o0 = 0.0f, o1 = 0.0f, o2 = 0.0f;
#pragma unroll
            for (int it = 0; it < 2; ++it) {
                float vx = bvx + o0, vy = bvy + o1, w = bw + o2;
                float af = theta - fast_atan2f(fmaf(w, 1.5f, vy), vx);
                float ar = fast_atan2f(fmaf(w, 1.4f, -vy), vx);

                // where(a>0, f(a), -f(-a)) == s * f(s*a), s = sign
                float sF = af > 0.0f ? 1.0f : -1.0f;
                float aF = sF * af, accF = fb2;
#pragma unroll
                for (int i = 0; i < 12; ++i) {
                    float h = fmaxf(fmaf(aF, w1f[i], b1f[i]), 0.0f);
                    accF = fmaf(h, w2f[i], accF);
                }
                float Ffy = sF * accF;

                float sR = ar > 0.0f ? 1.0f : -1.0f;
                float aR = sR * ar, accR = rb2;
#pragma unroll
                for (int i = 0; i < 12; ++i) {
                    float h = fmaxf(fmaf(aR, w1r[i], b1r[i]), 0.0f);
                    accR = fmaf(h, w2r[i], accR);
                }
                float Fry = sR * accR;

                float Frx = fmaf(rxw, vx * vx, a_pred + rxb);
                float vxd = Frx - Ffy * st + vy * w - 9.8f * sp;
                float vyd = Fry + Ffy * ct - vx * w;
                float wd  = 0.6325f * fmaf(Ffy * 1.5f, ct, -(Fry * 1.4f)); // MASS/IZ=0.6325
                o0 = fmaf(0.01f, vxd, o0);
                o1 = fmaf(0.01f, vyd, o1);
                o2 = fmaf(0.01f, wd,  o2);
            }
            if (s < B) *(F3*)(out + 3 * s) = F3{o0, o1, o2};   // global_store_b96
        }
        __syncthreads();   // protect buf before it is overwritten next iteration
    }
}

extern "C" void kernel_launch(void* const* d_in, const int* in_sizes, int n_in,
                              void* d_out, int out_size, void* d_ws, size_t ws_size,
                              hipStream_t stream) {
    const float* x     = (const float*)d_in[0];
    const float* fy_w1 = (const float*)d_in[1];
    const float* fy_b1 = (const float*)d_in[2];
    const float* fy_w2 = (const float*)d_in[3];
    const float* fy_b2 = (const float*)d_in[4];
    const float* ry_w1 = (const float*)d_in[5];
    const float* ry_b1 = (const float*)d_in[6];
    const float* ry_w2 = (const float*)d_in[7];
    const float* ry_b2 = (const float*)d_in[8];
    const float* rx_w  = (const float*)d_in[9];
    const float* rx_b  = (const float*)d_in[10];

    const int B = in_sizes[0] / 8;
    const int tiles = (B + TILE - 1) / TILE;
    const int tpb = 4;
    const int grid = (tiles + tpb - 1) / tpb;

    dyn_kernel<<<grid, TPB, 0, stream>>>(x, fy_w1, fy_b1, fy_w2, fy_b2,
                                         ry_w1, ry_b1, ry_w2, ry_b2,
                                         rx_w, rx_b, (float*)d_out, B, tpb);
}